// PostProcessModule_84215718740036
// MI455X (gfx1250) — compile-verified
//
#include <hip/hip_runtime.h>
#include <math.h>

// ---------------- problem constants (match reference) ----------------
#define NUM_IMGS    16
#define A_TOTAL     170280            // 90*160*9 + 45*80*9 + 23*40*9
#define L0_OFF      0
#define L1_OFF      129600
#define L2_OFF      162000
#define TOPK_C      1000
#define DETS        300
#define KCAND       2000              // 2 classes * 1000
#define KPAD        2048
#define SCORE_TH    0.05f
#define NMS_TH      0.5f
#define XFORM_CLIP  4.135166556742356f   // log(1000/16)
#define IMG_W       1280.0f
#define IMG_H       720.0f

// =====================================================================
// Kernel 1: sigmoid scores -> monotone uint32 keys laid out [b][c][A]
// Reads are fully coalesced over the cls tensors (B,18,fh,fw).
// key = float_bits(sigmoid(x)) if score > 0.05 else 0   (score in (0,1)
// so IEEE bits are order-preserving).
// =====================================================================
__global__ __launch_bounds__(256) void k_scores(
    const float* __restrict__ cls0, const float* __restrict__ cls1,
    const float* __restrict__ cls2, unsigned* __restrict__ keys) {
  const int S0 = NUM_IMGS * 18 * 14400;
  const int S1 = NUM_IMGS * 18 * 3600;
  const int S2 = NUM_IMGS * 18 * 920;
  int t = blockIdx.x * blockDim.x + threadIdx.x;
  if (t >= S0 + S1 + S2) return;

  const float* src; int fw, pc, levoff, local;
  if (t < S0)            { src = cls0; fw = 160; pc = 14400; levoff = L0_OFF; local = t; }
  else if (t < S0 + S1)  { src = cls1; fw = 80;  pc = 3600;  levoff = L1_OFF; local = t - S0; }
  else                   { src = cls2; fw = 40;  pc = 920;   levoff = L2_OFF; local = t - S0 - S1; }
  (void)fw;

  int per_b = 18 * pc;
  int b  = local / per_b;
  int r  = local - b * per_b;
  int ch = r / pc;                 // ch = k*2 + c
  int p  = r - ch * pc;            // pixel index y*fw + x
  int k  = ch >> 1;
  int c  = ch & 1;

  float v = src[local];
  float s = 1.0f / (1.0f + expf(-v));
  unsigned key = (s > SCORE_TH) ? __float_as_uint(s) : 0u;

  int anchor = levoff + p * 9 + k;
  keys[(size_t)(b * 2 + c) * A_TOTAL + anchor] = key;
}

// =====================================================================
// Kernel 2: exact top-1000 per (b,c) via 3-pass LDS radix select
// (11+11+10 bits of the key), then compaction. Exact-key ties resolved
// by smallest anchor index (reference jax.lax.top_k tie-break).
// One 1024-thread block per (b,c) => 32 blocks.
// =====================================================================
__global__ __launch_bounds__(1024) void k_select(
    const unsigned* __restrict__ keys, uint2* __restrict__ cand) {
  const int bc = blockIdx.x;
  const unsigned* kk = keys + (size_t)bc * A_TOTAL;
  __shared__ unsigned hist[2048];
  __shared__ unsigned eqbuf[1024];
  __shared__ unsigned sh_t1, sh_t2, sh_Kstar, sh_R, sh_cnt, sh_neq;
  const int tid = threadIdx.x;

  // ---- pass 1: top 11 bits ----
  hist[tid] = 0; hist[tid + 1024] = 0;
  __syncthreads();
  for (int a = tid; a < A_TOTAL; a += 1024) atomicAdd(&hist[kk[a] >> 21], 1u);
  __syncthreads();
  if (tid == 0) {
    unsigned cum = 0; int t = 0;
    for (int b = 2047; b >= 0; --b) {
      unsigned cc = hist[b];
      if (cum + cc >= (unsigned)TOPK_C || b == 0) { t = b; break; }
      cum += cc;
    }
    sh_t1 = (unsigned)t; sh_R = (unsigned)TOPK_C - cum;
  }
  __syncthreads();
  unsigned t1 = sh_t1;

  // ---- pass 2: middle 11 bits ----
  hist[tid] = 0; hist[tid + 1024] = 0;
  __syncthreads();
  for (int a = tid; a < A_TOTAL; a += 1024) {
    unsigned key = kk[a];
    if ((key >> 21) == t1) atomicAdd(&hist[(key >> 10) & 0x7FFu], 1u);
  }
  __syncthreads();
  if (tid == 0) {
    unsigned R = sh_R, cum = 0; int t = 0;
    for (int b = 2047; b >= 0; --b) {
      unsigned cc = hist[b];
      if (cum + cc >= R || b == 0) { t = b; break; }
      cum += cc;
    }
    sh_t2 = (unsigned)t; sh_R = R - cum;
  }
  __syncthreads();
  unsigned t2 = sh_t2;
  unsigned pre21 = (t1 << 11) | t2;

  // ---- pass 3: low 10 bits ----
  hist[tid] = 0; hist[tid + 1024] = 0;
  __syncthreads();
  for (int a = tid; a < A_TOTAL; a += 1024) {
    unsigned key = kk[a];
    if ((key >> 10) == pre21) atomicAdd(&hist[key & 0x3FFu], 1u);
  }
  __syncthreads();
  if (tid == 0) {
    unsigned R = sh_R, cum = 0; int t = 0;
    for (int b = 1023; b >= 0; --b) {
      unsigned cc = hist[b];
      if (cum + cc >= R || b == 0) { t = b; break; }
      cum += cc;
    }
    sh_Kstar = (t1 << 21) | (t2 << 10) | (unsigned)t;
    sh_R = R - cum;           // R3: number of exact-tie elements to take
    sh_cnt = 0; sh_neq = 0;
  }
  __syncthreads();
  unsigned Kstar = sh_Kstar, R3 = sh_R;

  // ---- compaction: all keys strictly greater than Kstar ----
  for (int a = tid; a < A_TOTAL; a += 1024) {
    unsigned key = kk[a];
    if (key > Kstar) {
      unsigned p = atomicAdd(&sh_cnt, 1u);
      if (p < (unsigned)TOPK_C) cand[(size_t)bc * TOPK_C + p] = make_uint2(key, (unsigned)a);
    }
  }
  __syncthreads();
  unsigned base = sh_cnt;   // == 1000 - R3

  if (R3 > 0) {
    if (Kstar == 0u) {
      // below-threshold padding: score -1 in reference => sentinel {0,0}
      for (unsigned s = base + tid; s < (unsigned)TOPK_C; s += 1024u)
        cand[(size_t)bc * TOPK_C + s] = make_uint2(0u, 0u);
    } else {
      // exact ties at Kstar: take R3 smallest anchor indices (deterministic)
      eqbuf[tid] = 0xFFFFFFFFu;
      __syncthreads();
      for (int a = tid; a < A_TOTAL; a += 1024) {
        if (kk[a] == Kstar) {
          unsigned p = atomicAdd(&sh_neq, 1u);
          if (p < 1024u) eqbuf[p] = (unsigned)a;
        }
      }
      __syncthreads();
      // in-LDS bitonic sort (ascending), 1024 elements, one per thread
      for (unsigned k2 = 2; k2 <= 1024; k2 <<= 1) {
        for (unsigned j = k2 >> 1; j > 0; j >>= 1) {
          unsigned i = (unsigned)tid, l = i ^ j;
          if (l > i) {
            unsigned a0 = eqbuf[i], a1 = eqbuf[l];
            bool up = ((i & k2) == 0);
            if ((a0 > a1) == up) { eqbuf[i] = a1; eqbuf[l] = a0; }
          }
          __syncthreads();
        }
      }
      for (unsigned r = tid; r < R3; r += 1024u)
        cand[(size_t)bc * TOPK_C + base + r] = make_uint2(Kstar, eqbuf[r]);
    }
  }
}

// =====================================================================
// Kernel 3: per-image merge-sort + decode + greedy NMS + top-300.
// One 1024-thread block (32 wave32s) per image. Candidate records are
// staged global->LDS with gfx1250 async loads (ASYNCcnt), everything
// else (sort keys, box coords, areas, flags, scan) stays in LDS.
// =====================================================================
__global__ __launch_bounds__(1024) void k_nms(
    const uint2* __restrict__ cand,
    const float* __restrict__ reg0, const float* __restrict__ reg1,
    const float* __restrict__ reg2, const float* __restrict__ anchors,
    float* __restrict__ outB, float* __restrict__ outS, int* __restrict__ outL) {
  const int b   = blockIdx.x;
  const int tid = threadIdx.x;

  __shared__ unsigned long long skey[KPAD];   // sort keys / staging (16KB)
  __shared__ float bx1[KPAD], by1[KPAD], bx2[KPAD], by2[KPAD];  // 32KB
  __shared__ unsigned areab[KPAD];            // area bits, reused as scan (8KB)
  __shared__ unsigned char removed[KPAD];     // 2KB

  // ---- stage 2000 candidate records into LDS via async global->LDS ----
  const unsigned long long* grec =
      (const unsigned long long*)(cand + (size_t)b * KCAND);
  for (int j = tid; j < KCAND; j += 1024) {
    unsigned lds_off = (unsigned)(size_t)(&skey[j]);
    unsigned long long gaddr = (unsigned long long)(size_t)(grec + j);
    asm volatile("global_load_async_to_lds_b64 %0, %1, off"
                 :: "v"(lds_off), "v"(gaddr) : "memory");
  }
  asm volatile("s_wait_asynccnt 0x0" ::: "memory");
  __syncthreads();

  // ---- build total-order 64-bit sort keys: (score bits desc, class asc,
  //      anchor asc).  low32 = ~((c<<18)|a).  Invalid (key==0) -> 0. ----
  for (int j = tid; j < KPAD; j += 1024) {
    unsigned long long sk = 0ull;
    if (j < KCAND) {
      unsigned long long rec = skey[j];
      unsigned key = (unsigned)rec;            // uint2.x (low dword)
      unsigned a   = (unsigned)(rec >> 32);    // uint2.y
      unsigned c   = (unsigned)(j / TOPK_C);   // cand layout [b][c][1000]
      if (key != 0u)
        sk = ((unsigned long long)key << 32) | (unsigned)(~((c << 18) | a));
    }
    skey[j] = sk;
  }
  __syncthreads();

  // ---- bitonic sort, descending, 2048 x u64 in LDS ----
  for (unsigned k2 = 2; k2 <= KPAD; k2 <<= 1) {
    for (unsigned j = k2 >> 1; j > 0; j >>= 1) {
      for (unsigned i = (unsigned)tid; i < KPAD; i += 1024u) {
        unsigned l = i ^ j;
        if (l > i) {
          unsigned long long a0 = skey[i], a1 = skey[l];
          bool up = ((i & k2) == 0);
          if (up ? (a0 < a1) : (a0 > a1)) { skey[i] = a1; skey[l] = a0; }
        }
      }
      __syncthreads();
    }
  }

  // ---- decode boxes (gather only the 2000 survivors) ----
  for (int s = tid; s < KPAD; s += 1024) {
    unsigned long long sk = skey[s];
    if (sk == 0ull) {
      bx1[s] = 0.f; by1[s] = 0.f; bx2[s] = 0.f; by2[s] = 0.f;
      areab[s] = __float_as_uint(0.f);
      removed[s] = 1;                       // ~valid
      continue;
    }
    unsigned v = ~((unsigned)sk);
    unsigned a = v & 0x3FFFFu;
    const float* reg; int off, fh, fw;
    if (a < (unsigned)L1_OFF)      { reg = reg0; off = L0_OFF; fh = 90; fw = 160; }
    else if (a < (unsigned)L2_OFF) { reg = reg1; off = L1_OFF; fh = 45; fw = 80; }
    else                           { reg = reg2; off = L2_OFF; fh = 23; fw = 40; }
    int idx = (int)a - off;
    int k   = idx % 9;
    int p   = idx / 9;
    int pc  = fh * fw;
    size_t base = ((size_t)b * 36 + (size_t)k * 4) * pc + p;
    float r0 = reg[base];
    float r1 = reg[base + (size_t)pc];
    float r2 = reg[base + (size_t)2 * pc];
    float r3 = reg[base + (size_t)3 * pc];

    float ax1 = anchors[4u * a], ay1 = anchors[4u * a + 1];
    float ax2 = anchors[4u * a + 2], ay2 = anchors[4u * a + 3];
    float w = ax2 - ax1, h = ay2 - ay1;
    float cx = ax1 + 0.5f * w, cy = ay1 + 0.5f * h;
    float dw = fminf(r2, XFORM_CLIP), dh = fminf(r3, XFORM_CLIP);
    float pcx = r0 * w + cx, pcy = r1 * h + cy;
    float pw = expf(dw) * w, ph = expf(dh) * h;
    float X1 = fminf(fmaxf(pcx - 0.5f * pw, 0.f), IMG_W);
    float Y1 = fminf(fmaxf(pcy - 0.5f * ph, 0.f), IMG_H);
    float X2 = fminf(fmaxf(pcx + 0.5f * pw, 0.f), IMG_W);
    float Y2 = fminf(fmaxf(pcy + 0.5f * ph, 0.f), IMG_H);
    bx1[s] = X1; by1[s] = Y1; bx2[s] = X2; by2[s] = Y2;
    areab[s] = __float_as_uint((X2 - X1) * (Y2 - Y1));
    removed[s] = 0;
  }
  __syncthreads();

  // ---- greedy NMS: exact reference fori_loop semantics ----
  for (int i = 0; i < KCAND; ++i) {
    if (!removed[i]) {   // uniform branch (all threads read same LDS byte)
      float xi1 = bx1[i], yi1 = by1[i], xi2 = bx2[i], yi2 = by2[i];
      float ai  = __uint_as_float(areab[i]);
      for (int j = i + 1 + tid; j < KCAND; j += 1024) {
        if (!removed[j]) {
          float xx1 = fmaxf(xi1, bx1[j]);
          float yy1 = fmaxf(yi1, by1[j]);
          float xx2 = fminf(xi2, bx2[j]);
          float yy2 = fminf(yi2, by2[j]);
          float inter = fmaxf(xx2 - xx1, 0.f) * fmaxf(yy2 - yy1, 0.f);
          float uni = ai + __uint_as_float(areab[j]) - inter;
          float iou = (uni > 0.f) ? (inter / uni) : 0.f;
          if (iou > NMS_TH) removed[j] = 1;
        }
      }
    }
    __syncthreads();
  }

  // ---- stable compaction (inclusive Hillis-Steele scan, reuse areab) ----
  for (int s = tid; s < KPAD; s += 1024) areab[s] = removed[s] ? 0u : 1u;
  __syncthreads();
  for (unsigned d = 1; d < KPAD; d <<= 1) {
    unsigned i0 = (unsigned)tid, i1 = (unsigned)tid + 1024u;
    unsigned v0 = (i0 >= d) ? areab[i0 - d] : 0u;
    unsigned v1 = (i1 >= d) ? areab[i1 - d] : 0u;
    __syncthreads();
    areab[i0] += v0; areab[i1] += v1;
    __syncthreads();
  }
  unsigned nkept = areab[KPAD - 1];

  // ---- write top-300 per image (sorted order == top_k order) ----
  for (int s = tid; s < KPAD; s += 1024) {
    if (!removed[s]) {
      unsigned pos = areab[s] - 1u;   // exclusive rank among kept
      if (pos < (unsigned)DETS) {
        unsigned long long sk = skey[s];
        unsigned key = (unsigned)(sk >> 32);
        unsigned v   = ~((unsigned)sk);
        int c = (int)((v >> 18) & 1u);
        size_t o = (size_t)(b * DETS + (int)pos);
        outB[o * 4 + 0] = bx1[s];
        outB[o * 4 + 1] = by1[s];
        outB[o * 4 + 2] = bx2[s];
        outB[o * 4 + 3] = by2[s];
        outS[o] = __uint_as_float(key);
        outL[o] = c;
      }
    }
  }
  unsigned nk = nkept < (unsigned)DETS ? nkept : (unsigned)DETS;
  for (unsigned p = nk + (unsigned)tid; p < (unsigned)DETS; p += 1024u) {
    size_t o = (size_t)b * DETS + p;
    outB[o * 4 + 0] = 0.f; outB[o * 4 + 1] = 0.f;
    outB[o * 4 + 2] = 0.f; outB[o * 4 + 3] = 0.f;
    outS[o] = 0.f;
    outL[o] = -1;
  }
}

// =====================================================================
extern "C" void kernel_launch(void* const* d_in, const int* in_sizes, int n_in,
                              void* d_out, int out_size, void* d_ws, size_t ws_size,
                              hipStream_t stream) {
  (void)in_sizes; (void)n_in; (void)out_size; (void)ws_size;
  const float* reg0    = (const float*)d_in[0];
  const float* cls0    = (const float*)d_in[1];
  const float* reg1    = (const float*)d_in[2];
  const float* cls1    = (const float*)d_in[3];
  const float* reg2    = (const float*)d_in[4];
  const float* cls2    = (const float*)d_in[5];
  const float* anchors = (const float*)d_in[6];

  unsigned* keys = (unsigned*)d_ws;                               // 21.8 MB
  uint2*    cand = (uint2*)((char*)d_ws +
                            (size_t)NUM_IMGS * 2 * A_TOTAL * sizeof(unsigned));

  float* outB = (float*)d_out;                                    // [16][300][4]
  float* outS = outB + (size_t)NUM_IMGS * DETS * 4;               // [16][300]
  int*   outL = (int*)(outS + (size_t)NUM_IMGS * DETS);           // [16][300] i32

  int n1 = NUM_IMGS * 18 * (14400 + 3600 + 920);
  k_scores<<<(n1 + 255) / 256, 256, 0, stream>>>(cls0, cls1, cls2, keys);
  k_select<<<NUM_IMGS * 2, 1024, 0, stream>>>(keys, cand);
  k_nms<<<NUM_IMGS, 1024, 0, stream>>>(cand, reg0, reg1, reg2, anchors,
                                       outB, outS, outL);
}